// Dual_Loss_85229331022461
// MI455X (gfx1250) — compile-verified
//
#include <hip/hip_runtime.h>
#include <cmath>

typedef float v4f __attribute__((ext_vector_type(4)));
typedef int   v4i __attribute__((vector_size(16)));   // matches builtin's param type

#define AS1 __attribute__((address_space(1)))
#define AS3 __attribute__((address_space(3)))

#if defined(__gfx1250__) && __has_builtin(__builtin_amdgcn_global_load_async_to_lds_b128) && \
    __has_builtin(__builtin_amdgcn_s_wait_asynccnt)
#define USE_ASYNC_LDS 1
#else
#define USE_ASYNC_LDS 0
#endif

constexpr int B_ROWS = 16384;
constexpr int C_COLS = 4096;
constexpr int TPB    = 256;                 // 8 wave32s per block
constexpr int CHUNKS = C_COLS / (TPB * 4);  // float4 chunks per thread = 4

// One block per row. Stage p-row and pai-row in LDS via async copies, then
// exact two-pass softmax-CE: pass1 row max, pass2 sum-exp + dot(pai,p) + sum(pai).
__global__ __launch_bounds__(TPB) void row_ce_kernel(const float* __restrict__ p,
                                                     const float* __restrict__ pai,
                                                     float* __restrict__ ce_rows) {
    __shared__ __align__(16) float lds_p[C_COLS];
    __shared__ __align__(16) float lds_q[C_COLS];
    __shared__ float red_m[8];
    __shared__ float red_s[8];
    __shared__ float red_d[8];
    __shared__ float red_sp[8];

    const int row  = blockIdx.x;
    const int tid  = threadIdx.x;
    const int lane = tid & 31;
    const int wid  = tid >> 5;

    const float* prow = p   + (size_t)row * C_COLS;
    const float* qrow = pai + (size_t)row * C_COLS;

#if USE_ASYNC_LDS
    {
        AS1 v4i* gp = (AS1 v4i*)prow;       // const cast away: copy source only
        AS1 v4i* gq = (AS1 v4i*)qrow;
        AS3 v4i* lp = (AS3 v4i*)lds_p;
        AS3 v4i* lq = (AS3 v4i*)lds_q;
#pragma unroll
        for (int k = 0; k < CHUNKS; ++k) {
            const int idx = k * TPB + tid;  // 16-byte element index
            __builtin_amdgcn_global_load_async_to_lds_b128(gp + idx, lp + idx, 0, 0);
            __builtin_amdgcn_global_load_async_to_lds_b128(gq + idx, lq + idx, 0, 0);
        }
        __builtin_amdgcn_s_wait_asynccnt(0);
    }
    __syncthreads();
#else
#pragma unroll
    for (int k = 0; k < CHUNKS; ++k) {
        const int c = (k * TPB + tid) * 4;
        *(v4f*)&lds_p[c] = *(const v4f*)&prow[c];
        *(v4f*)&lds_q[c] = *(const v4f*)&qrow[c];
    }
    __syncthreads();
#endif

    // ---- pass 1: row max (matches jax log_softmax max-subtraction) ----
    float m = -INFINITY;
#pragma unroll
    for (int k = 0; k < CHUNKS; ++k) {
        const v4f pv = *(const v4f*)&lds_p[(k * TPB + tid) * 4];
        m = fmaxf(m, fmaxf(fmaxf(pv.x, pv.y), fmaxf(pv.z, pv.w)));
    }
#pragma unroll
    for (int off = 16; off > 0; off >>= 1) m = fmaxf(m, __shfl_xor(m, off, 32));
    if (lane == 0) red_m[wid] = m;
    __syncthreads();
    float M = red_m[0];
#pragma unroll
    for (int i = 1; i < 8; ++i) M = fmaxf(M, red_m[i]);

    // ---- pass 2: S = sum exp(p-M), D = dot(pai,p), SP = sum pai ----
    float s = 0.0f, d = 0.0f, sp = 0.0f;
#pragma unroll
    for (int k = 0; k < CHUNKS; ++k) {
        const int c = (k * TPB + tid) * 4;
        const v4f pv = *(const v4f*)&lds_p[c];
        const v4f qv = *(const v4f*)&lds_q[c];
        s += __expf(pv.x - M); d += qv.x * pv.x; sp += qv.x;
        s += __expf(pv.y - M); d += qv.y * pv.y; sp += qv.y;
        s += __expf(pv.z - M); d += qv.z * pv.z; sp += qv.z;
        s += __expf(pv.w - M); d += qv.w * pv.w; sp += qv.w;
    }
#pragma unroll
    for (int off = 16; off > 0; off >>= 1) {
        s  += __shfl_xor(s,  off, 32);
        d  += __shfl_xor(d,  off, 32);
        sp += __shfl_xor(sp, off, 32);
    }
    if (lane == 0) { red_s[wid] = s; red_d[wid] = d; red_sp[wid] = sp; }
    __syncthreads();

    if (tid == 0) {
        float S = 0.0f, D = 0.0f, SP = 0.0f;
#pragma unroll
        for (int i = 0; i < 8; ++i) { S += red_s[i]; D += red_d[i]; SP += red_sp[i]; }
        // ce_row = -sum_j pai*(p - M - log S) = (M + log S)*SP - D
        ce_rows[row] = (M + __logf(S)) * SP - D;
    }
}

// Single-block deterministic final reduction: scalar = mse/B + ce_total/20.
__global__ __launch_bounds__(1024) void final_reduce_kernel(const float* __restrict__ ce_rows,
                                                            const float* __restrict__ v,
                                                            const float* __restrict__ z,
                                                            float* __restrict__ out) {
    __shared__ float red_ce[32];
    __shared__ float red_ms[32];
    const int tid  = threadIdx.x;
    const int lane = tid & 31;
    const int wid  = tid >> 5;

    float ace = 0.0f, amse = 0.0f;
    for (int i = tid; i < B_ROWS; i += 1024) {
        ace += ce_rows[i];
        const float dv = z[i] - v[i];
        amse = fmaf(dv, dv, amse);
    }
#pragma unroll
    for (int off = 16; off > 0; off >>= 1) {
        ace  += __shfl_xor(ace,  off, 32);
        amse += __shfl_xor(amse, off, 32);
    }
    if (lane == 0) { red_ce[wid] = ace; red_ms[wid] = amse; }
    __syncthreads();
    if (tid == 0) {
        float S = 0.0f, Mv = 0.0f;
#pragma unroll
        for (int i = 0; i < 32; ++i) { S += red_ce[i]; Mv += red_ms[i]; }
        out[0] = Mv / (float)B_ROWS + S / 20.0f;
    }
}

extern "C" void kernel_launch(void* const* d_in, const int* in_sizes, int n_in,
                              void* d_out, int out_size, void* d_ws, size_t ws_size,
                              hipStream_t stream) {
    (void)in_sizes; (void)n_in; (void)out_size; (void)ws_size;
    const float* p   = (const float*)d_in[0];
    const float* v   = (const float*)d_in[1];
    const float* z   = (const float*)d_in[2];
    const float* pai = (const float*)d_in[3];
    float* ce_rows   = (float*)d_ws;   // 16384 floats = 64 KB scratch

    row_ce_kernel<<<B_ROWS, TPB, 0, stream>>>(p, pai, ce_rows);
    final_reduce_kernel<<<1, 1024, 0, stream>>>(ce_rows, v, z, (float*)d_out);
}